// PairTabModel_50233937494467
// MI455X (gfx1250) — compile-verified
//
#include <hip/hip_runtime.h>
#include <hip/hip_bf16.h>
#include <stdint.h>

// Problem constants (from the reference)
#define NFRAMES 8
#define NLOC    4096
#define NALL    8192
#define NNEI    256
#define NTYPES  4
#define NSPLINE 1024
#define RMIN_F  0.0f
#define HH_F    0.02f

#define ATOMS_PER_BLOCK 8        // one wave32 per atom
#define BLOCK_THREADS   256      // 8 waves
#define BLOCKS_PER_FRAME (NLOC / ATOMS_PER_BLOCK)   // 512

__global__ __launch_bounds__(BLOCK_THREADS)
void pair_tab_energy_kernel(const float* __restrict__ coord,   // [NF, NALL, 3]
                            const int*   __restrict__ atype,   // [NF, NALL]
                            const int*   __restrict__ nlist,   // [NF, NLOC, NNEI]
                            const float* __restrict__ tab,     // [NT, NT, NSPLINE, 4]
                            float*       __restrict__ out)     // [NF, NLOC]
{
    // nlist tile for this block: 8 atoms x 256 neighbors x 4B = 8 KB
    __shared__ __align__(16) int snl[ATOMS_PER_BLOCK * NNEI];

    const int tid      = threadIdx.x;
    const int frame    = blockIdx.x / BLOCKS_PER_FRAME;
    const int atomBase = (blockIdx.x % BLOCKS_PER_FRAME) * ATOMS_PER_BLOCK;

    // ---- Async-stage the (coalesced) nlist tile into LDS (CDNA5 async path) ----
    // 8 KB = 512 x 16B chunks; 256 threads issue 2 chunks each.
    {
        const unsigned long long gbase =
            (unsigned long long)(uintptr_t)(nlist + ((size_t)frame * NLOC + atomBase) * NNEI);
        // Low 32 bits of a flat shared-aperture address are the LDS byte offset
        // (ISA 10.2).
        const unsigned ldsbase = (unsigned)(uintptr_t)&snl[0];
#pragma unroll
        for (int c = 0; c < 2; ++c) {
            unsigned byteoff = (unsigned)(tid + c * BLOCK_THREADS) * 16u;
            unsigned ldsaddr = ldsbase + byteoff;
            // GVS mode: mem_addr = SGPR64 + VGPR32 offset; VDST = LDS byte address.
            asm volatile("global_load_async_to_lds_b128 %0, %1, %2"
                         :
                         : "v"(ldsaddr), "v"(byteoff), "s"(gbase)
                         : "memory");
        }
        asm volatile("s_wait_asynccnt 0x0" ::: "memory");
    }
    __syncthreads();

    // ---- One wave32 per local atom ----
    const int wave = tid >> 5;   // 0..7
    const int lane = tid & 31;   // 0..31
    const int ia   = atomBase + wave;

    const size_t fa_i = (size_t)frame * NALL + (size_t)ia;
    const float  xi   = coord[fa_i * 3 + 0];
    const float  yi   = coord[fa_i * 3 + 1];
    const float  zi   = coord[fa_i * 3 + 2];
    const int    ti   = atype[fa_i];

    const float* tab_i = tab + (size_t)ti * NTYPES * NSPLINE * 4;

    float acc = 0.0f;

#pragma unroll
    for (int k = 0; k < NNEI / 32; ++k) {
        const int j     = snl[wave * NNEI + k * 32 + lane];
        const bool valid = (j != -1);
        const int jm    = valid ? j : 0;

        const size_t fa_j = (size_t)frame * NALL + (size_t)jm;
        const float dx = xi - coord[fa_j * 3 + 0];
        const float dy = yi - coord[fa_j * 3 + 1];
        const float dz = zi - coord[fa_j * 3 + 2];
        const float rr = sqrtf(dx * dx + dy * dy + dz * dz);
        const int   tj = atype[fa_j];

        // uu >= 0 always (rr >= 0, RMIN = 0), so idx >= 0 and clip(idx,0,..)
        // is a no-op on the path where coefficients are nonzero.
        const float uu   = (rr - RMIN_F) / HH_F;
        const int   idx  = (int)uu;            // truncation == astype(int32)
        const float frac = uu - (float)idx;

        if (valid && idx < NSPLINE) {
            const float4 c =
                *(const float4*)(tab_i + ((size_t)tj * NSPLINE + (size_t)idx) * 4);
            acc += ((c.w * frac + c.z) * frac + c.y) * frac + c.x;
        }
        // invalid or idx >= NSPLINE: reference zeroes the coefficients -> 0 contribution
    }

    // wave32 butterfly reduction
#pragma unroll
    for (int s = 16; s > 0; s >>= 1)
        acc += __shfl_xor(acc, s, 32);

    if (lane == 0)
        out[(size_t)frame * NLOC + (size_t)ia] = 0.5f * acc;
}

extern "C" void kernel_launch(void* const* d_in, const int* in_sizes, int n_in,
                              void* d_out, int out_size, void* d_ws, size_t ws_size,
                              hipStream_t stream)
{
    const float* coord = (const float*)d_in[0];   // extended_coord
    const int*   atype = (const int*)  d_in[1];   // extended_atype
    const int*   nlist = (const int*)  d_in[2];   // nlist
    const float* tab   = (const float*)d_in[3];   // tab_data
    float*       out   = (float*)d_out;           // [NFRAMES, NLOC]

    (void)in_sizes; (void)n_in; (void)out_size; (void)d_ws; (void)ws_size;

    dim3 grid(NFRAMES * BLOCKS_PER_FRAME);
    dim3 block(BLOCK_THREADS);
    pair_tab_energy_kernel<<<grid, block, 0, stream>>>(coord, atype, nlist, tab, out);
}